// Cheby_49323404427972
// MI455X (gfx1250) — compile-verified
//
#include <hip/hip_runtime.h>
#include <hip/hip_bf16.h>
#include <cstdint>

#define NNODES 100000
#define NEDGES 1600000
#define FDIM   64
#define FOUT   40

typedef __attribute__((ext_vector_type(2))) float v2f;
typedef __attribute__((ext_vector_type(4))) float v4f;
typedef __attribute__((ext_vector_type(8))) float v8f;

// ---------------- degree: deg[row[e]] += w[e] ----------------
__global__ void deg_kernel(const int* __restrict__ ei, const float* __restrict__ w,
                           float* __restrict__ deg, int E) {
    int e = blockIdx.x * blockDim.x + threadIdx.x;
    if (e < E) unsafeAtomicAdd(&deg[ei[e]], w[e]);
}

// ---------------- norm[e] = -dinv[row]*w*dinv[col] ----------------
__global__ void norm_kernel(const int* __restrict__ ei, const float* __restrict__ w,
                            const float* __restrict__ deg, float* __restrict__ norm, int E) {
    int e = blockIdx.x * blockDim.x + threadIdx.x;
    if (e < E) {
        int r = ei[e];
        int c = ei[E + e];
        float dr = deg[r], dc = deg[c];
        float ir = dr > 0.0f ? rsqrtf(dr) : 0.0f;
        float ic = dc > 0.0f ? rsqrtf(dc) : 0.0f;
        norm[e] = -ir * w[e] * ic;
    }
}

// ---------------- out4[i] = -in4[i] (vectorized), used to seed Tx2 = -Tx0 ----------------
__global__ void neg_copy_kernel(const float* __restrict__ in, float* __restrict__ out, int n4) {
    int i = blockIdx.x * blockDim.x + threadIdx.x;
    if (i < n4) {
        v4f v = ((const v4f*)in)[i];
        ((v4f*)out)[i] = -v;
    }
}

// ---------------- SpMM scatter: out[row] += scale*norm[e]*h[col], 4 feats/thread ----------------
__global__ void spmm_kernel(const float* __restrict__ h, const int* __restrict__ ei,
                            const float* __restrict__ norm, float scale,
                            float* __restrict__ out, int E) {
    int tid = blockIdx.x * blockDim.x + threadIdx.x;
    int e = tid >> 4;          // 16 threads per edge (64 feats / 4)
    if (e >= E) return;
    int q = (tid & 15) << 2;   // feature base
    int r = ei[e];
    int c = ei[E + e];
    float s = scale * norm[e];
    v4f v = *(const v4f*)(h + (size_t)c * FDIM + q);
    float* op = out + (size_t)r * FDIM + q;
    unsafeAtomicAdd(op + 0, s * v.x);
    unsafeAtomicAdd(op + 1, s * v.y);
    unsafeAtomicAdd(op + 2, s * v.z);
    unsafeAtomicAdd(op + 3, s * v.w);
}

// ---------------- fused dense part: out = relu(H0*W[0] + T1*W[1] + T2*W[2] + b) ----------------
// One wave32 computes a full 16 x OUTFP row-block with V_WMMA_F32_16X16X4_F32.
// W is staged in LDS, zero-padded to OUTFP columns and pair-interleaved so a B
// fragment is one ds_load_b64 with compile-time strides.
template <int OUTF, int OUTFP>
__global__ void __launch_bounds__(256) cheb_gemm_kernel(
    const float* __restrict__ H0, const float* __restrict__ T1, const float* __restrict__ T2,
    const float* __restrict__ W,    // [3][64][OUTF] row-major in global
    const float* __restrict__ bias,
    float* __restrict__ out, int mTiles) {

    constexpr int NT = OUTFP / 16;                 // column tiles per row-block
    __shared__ float Wl[3 * 32 * OUTFP * 2];       // [p][k/2][j][k&1]

    // Cooperative stage: pad columns >= OUTF with zeros, interleave (k, k+1) pairs.
    for (int idx = threadIdx.x; idx < 3 * 64 * OUTFP; idx += 256) {
        int p   = idx / (64 * OUTFP);
        int rem = idx - p * 64 * OUTFP;
        int k   = rem / OUTFP;
        int j   = rem - k * OUTFP;
        float v = (j < OUTF) ? W[((size_t)p * 64 + k) * OUTF + j] : 0.0f;
        Wl[((p * 32 + (k >> 1)) * OUTFP + j) * 2 + (k & 1)] = v;
    }
    __syncthreads();

    int wave = blockIdx.x * 8 + (threadIdx.x >> 5);
    if (wave >= mTiles) return;                    // wave-uniform: EXEC all-1s for WMMA

    int lane = threadIdx.x & 31;
    int half = lane >> 4;                          // 0: lanes 0-15, 1: lanes 16-31
    int m    = lane & 15;
    size_t rowA = (size_t)wave * 16 + m;           // A row for this lane

    v8f acc[NT] = {};

    const float* Hs[3] = {H0, T1, T2};
#pragma unroll
    for (int p = 0; p < 3; ++p) {
        const float* __restrict__ arow = Hs[p] + rowA * FDIM + 2 * half;
        const float* __restrict__ Wp   = Wl + p * (32 * OUTFP * 2);
#pragma unroll
        for (int k0 = 0; k0 < FDIM; k0 += 4) {
            // A fragment (16x4 f32): lane m holds K = k0+2*half, k0+2*half+1
            v2f a = *(const v2f*)(arow + k0);
            int kp = (k0 >> 1) + half;             // pair index: kb = 2*kp
#pragma unroll
            for (int t = 0; t < NT; ++t) {
                // B fragment: lanes 0-15 hold K=k0..k0+1, lanes 16-31 K=k0+2..k0+3
                v2f b = *(const v2f*)(Wp + (kp * OUTFP + t * 16 + m) * 2);
                acc[t] = __builtin_amdgcn_wmma_f32_16x16x4_f32(
                             false, a, false, b, (short)0, acc[t], false, false);
            }
        }
    }

    // C/D layout: VGPR v -> M = v + 8*half, N = lane&15
#pragma unroll
    for (int t = 0; t < NT; ++t) {
        int colN = t * 16 + m;
        if (colN < OUTF) {
            float bval = bias[colN];
#pragma unroll
            for (int v = 0; v < 8; ++v) {
                int r = wave * 16 + v + 8 * half;
                out[(size_t)r * OUTF + colN] = fmaxf(acc[t][v] + bval, 0.0f);
            }
        }
    }
}

// ---------------- log_softmax over last dim (40) ----------------
__global__ void lsm_kernel(const float* __restrict__ in, float* __restrict__ out, int n, int f) {
    int i = blockIdx.x * blockDim.x + threadIdx.x;
    if (i >= n) return;
    const float* p = in + (size_t)i * f;
    float mx = -3.4e38f;
    for (int j = 0; j < f; ++j) mx = fmaxf(mx, p[j]);
    float s = 0.0f;
    for (int j = 0; j < f; ++j) s += __expf(p[j] - mx);
    float lse = mx + __logf(s);
    float* o = out + (size_t)i * f;
    for (int j = 0; j < f; ++j) o[j] = p[j] - lse;
}

// ---------------- one ChebConv layer ----------------
template <int OUTF, int OUTFP>
static inline void run_layer(const float* Hin, const float* W, const float* b,
                             float* T1, float* T2, float* Hout,
                             const int* ei, const float* norm, hipStream_t stream) {
    const int spmmThreads = NEDGES * 16;
    hipMemsetAsync(T1, 0, (size_t)NNODES * FDIM * sizeof(float), stream);
    // T1 = L * Hin
    spmm_kernel<<<(spmmThreads + 255) / 256, 256, 0, stream>>>(Hin, ei, norm, 1.0f, T1, NEDGES);
    // T2 = -Hin, then T2 += 2 * L * T1
    neg_copy_kernel<<<(NNODES * 16 + 255) / 256, 256, 0, stream>>>(Hin, T2, NNODES * 16);
    spmm_kernel<<<(spmmThreads + 255) / 256, 256, 0, stream>>>(T1, ei, norm, 2.0f, T2, NEDGES);
    // Hout = relu(Hin*W0 + T1*W1 + T2*W2 + b)
    const int mTiles = NNODES / 16;                // 100000 % 16 == 0
    cheb_gemm_kernel<OUTF, OUTFP><<<(mTiles + 7) / 8, 256, 0, stream>>>(
        Hin, T1, T2, W, b, Hout, mTiles);
}

extern "C" void kernel_launch(void* const* d_in, const int* in_sizes, int n_in,
                              void* d_out, int out_size, void* d_ws, size_t ws_size,
                              hipStream_t stream) {
    (void)in_sizes; (void)n_in; (void)out_size; (void)ws_size;

    const float* x  = (const float*)d_in[0];
    const int*   ei = (const int*)d_in[1];     // [2][E]
    const float* ea = (const float*)d_in[2];
    const float* W0 = (const float*)d_in[3];
    const float* b0 = (const float*)d_in[4];
    const float* W1 = (const float*)d_in[5];
    const float* b1 = (const float*)d_in[6];
    const float* W2 = (const float*)d_in[7];
    const float* b2 = (const float*)d_in[8];
    float* out = (float*)d_out;

    // workspace carve-out (floats)
    float* ws   = (float*)d_ws;
    float* deg  = ws;                               // N
    float* norm = deg + NNODES;                     // E
    float* T1   = norm + NEDGES;                    // N*64
    float* T2   = T1 + (size_t)NNODES * FDIM;       // N*64
    float* HA   = T2 + (size_t)NNODES * FDIM;       // N*64
    float* HB   = HA + (size_t)NNODES * FDIM;       // N*64

    // ---- normalization ----
    hipMemsetAsync(deg, 0, (size_t)NNODES * sizeof(float), stream);
    deg_kernel<<<(NEDGES + 255) / 256, 256, 0, stream>>>(ei, ea, deg, NEDGES);
    norm_kernel<<<(NEDGES + 255) / 256, 256, 0, stream>>>(ei, ea, deg, norm, NEDGES);

    // ---- three ChebConv + ReLU layers ----
    run_layer<FDIM, 64>(x,  W0, b0, T1, T2, HA, ei, norm, stream);
    run_layer<FDIM, 64>(HA, W1, b1, T1, T2, HB, ei, norm, stream);
    run_layer<FOUT, 48>(HB, W2, b2, T1, T2, HA, ei, norm, stream);

    // ---- log_softmax -> d_out ----
    lsm_kernel<<<(NNODES + 255) / 256, 256, 0, stream>>>(HA, out, NNODES, FOUT);
}